// SimilarityModuleAINET_37134287241507
// MI455X (gfx1250) — compile-verified
//
#include <hip/hip_runtime.h>
#include <hip/hip_bf16.h>

#define NSPIX 256
#define FEAT  64

typedef __attribute__((ext_vector_type(2))) float v2f;
typedef __attribute__((ext_vector_type(8))) float v8f;

// ---------------------------------------------------------------------------
// Kernel 0: zero the moment accumulators in workspace
// ---------------------------------------------------------------------------
__global__ void zero_kernel(float* __restrict__ p, int n) {
    int i = blockIdx.x * blockDim.x + threadIdx.x;
    if (i < n) p[i] = 0.0f;
}

// ---------------------------------------------------------------------------
// Kernel 1: per-pixel 10-moment segment accumulation.
//   moments per (b, s): [count, Sx0, Sx1, Sx2, Sx00, Sx01, Sx02, Sx11, Sx12, Sx22]
// LDS accumulators (256*10 floats = 10KB) per block, global atomic flush.
// ---------------------------------------------------------------------------
__global__ void __launch_bounds__(256) moments_kernel(
        const float* __restrict__ x, const int* __restrict__ labels,
        float* __restrict__ mom, int HW) {
    __shared__ float smom[NSPIX * 10];
    const int b = blockIdx.y;
    for (int i = threadIdx.x; i < NSPIX * 10; i += blockDim.x) smom[i] = 0.0f;
    __syncthreads();

    const float* x0 = x + (size_t)b * 3 * HW;
    const float* x1 = x0 + HW;
    const float* x2 = x1 + HW;
    const int* lab  = labels + (size_t)b * HW;

    const int stride = blockDim.x * gridDim.x;
    for (int p = blockIdx.x * blockDim.x + threadIdx.x; p < HW; p += stride) {
        float a = x0[p], c = x1[p], d = x2[p];
        int s = lab[p];
        float* m = smom + s * 10;
        atomicAdd(m + 0, 1.0f);
        atomicAdd(m + 1, a);
        atomicAdd(m + 2, c);
        atomicAdd(m + 3, d);
        atomicAdd(m + 4, a * a);
        atomicAdd(m + 5, a * c);
        atomicAdd(m + 6, a * d);
        atomicAdd(m + 7, c * c);
        atomicAdd(m + 8, c * d);
        atomicAdd(m + 9, d * d);
    }
    __syncthreads();

    float* gm = mom + (size_t)b * NSPIX * 10;
    for (int i = threadIdx.x; i < NSPIX * 10; i += blockDim.x) {
        float v = smom[i];
        if (v != 0.0f) atomicAdd(gm + i, v);
    }
}

// ---------------------------------------------------------------------------
// Kernel 2: per-segment reconstruction (64 threads per segment).
//   mean_f = (W_pix[f]·Sx + b_f*n) / max(n,1)
//   E[f^2] = (w M w^T + 2 b_f (w·Sx) + b_f^2 n) / max(n,1)
//   std = sqrt(max(var,1e-6)); reduced = W_red @ [mean;std] + b_red; L2-normalize
// ---------------------------------------------------------------------------
__global__ void __launch_bounds__(64) segment_stats_kernel(
        const float* __restrict__ mom,
        const float* __restrict__ W_pix, const float* __restrict__ b_pix,
        const float* __restrict__ W_red, const float* __restrict__ b_red,
        float* __restrict__ rn) {
    const int seg = blockIdx.x;       // 0 .. B*NSPIX-1
    const int f   = threadIdx.x;      // 0 .. 63

    __shared__ float m10[10];
    __shared__ float comb[2 * FEAT];
    __shared__ float red[FEAT];
    __shared__ float sq[FEAT];

    if (f < 10) m10[f] = mom[(size_t)seg * 10 + f];
    __syncthreads();

    const float n    = m10[0];
    const float invc = 1.0f / fmaxf(n, 1.0f);
    const float sx = m10[1], sy = m10[2], sz = m10[3];

    const float w0 = W_pix[f * 3 + 0];
    const float w1 = W_pix[f * 3 + 1];
    const float w2 = W_pix[f * 3 + 2];
    const float bf = b_pix[f];

    const float dot  = w0 * sx + w1 * sy + w2 * sz;
    const float mean = (dot + bf * n) * invc;
    const float q = w0 * w0 * m10[4] + 2.0f * w0 * w1 * m10[5] + 2.0f * w0 * w2 * m10[6]
                  + w1 * w1 * m10[7] + 2.0f * w1 * w2 * m10[8] + w2 * w2 * m10[9];
    const float ef2 = (q + 2.0f * bf * dot + bf * bf * n) * invc;
    const float var = ef2 - mean * mean;
    const float sd  = sqrtf(fmaxf(var, 1e-6f));

    comb[f]        = mean;
    comb[FEAT + f] = sd;
    __syncthreads();

    float r = b_red[f];
    const float* wr = W_red + (size_t)f * (2 * FEAT);
#pragma unroll 8
    for (int g = 0; g < 2 * FEAT; ++g) r += wr[g] * comb[g];
    red[f] = r;
    sq[f]  = r * r;
    __syncthreads();

    // tree-reduce sum of squares over 64 threads
    for (int off = 32; off > 0; off >>= 1) {
        if (f < off) sq[f] += sq[f + off];
        __syncthreads();
    }
    const float nrm = fmaxf(sqrtf(sq[0]), 1e-12f);
    rn[(size_t)seg * FEAT + f] = red[f] / nrm;
}

// ---------------------------------------------------------------------------
// Kernel 3: sim = rn @ rn^T per batch via V_WMMA_F32_16X16X4_F32.
// One wave32 per 16x16 output tile; 16 WMMA ops over K=64.
// Fused epilogue: y = relu(scale*sim + shift) folding conv1x1 + eval-BN.
// A-fragment layout (16x4 f32): vgpr k%2, lane (k/2)*16 + m.
// C/D layout (16x16 f32): VGPR v, lane -> row = v + 8*(lane>>4), col = lane&15.
// ---------------------------------------------------------------------------
__global__ void __launch_bounds__(256) sim_wmma_kernel(
        const float* __restrict__ rn,
        const float* __restrict__ w_sim, const float* __restrict__ b_sim,
        const float* __restrict__ bn_g, const float* __restrict__ bn_b,
        const float* __restrict__ bn_m, const float* __restrict__ bn_v,
        float* __restrict__ out) {
    const int gt   = blockIdx.x * blockDim.x + threadIdx.x;
    const int wave = gt >> 5;            // 0 .. B*16*16-1
    const int lane = threadIdx.x & 31;

    const int b  = wave >> 8;
    const int tm = (wave >> 4) & 15;
    const int tn = wave & 15;

    const float* base = rn + (size_t)b * NSPIX * FEAT;
    const int half = lane >> 4;          // 0: K pair {0,1}; 1: K pair {2,3}
    const int l    = lane & 15;
    const float* rowA = base + (size_t)(tm * 16 + l) * FEAT;
    const float* rowB = base + (size_t)(tn * 16 + l) * FEAT;

    v8f acc = {};
#pragma unroll
    for (int k0 = 0; k0 < FEAT; k0 += 4) {
        const int k = k0 + 2 * half;
        v2f a  = *(const v2f*)(rowA + k);
        v2f bm = *(const v2f*)(rowB + k);
        acc = __builtin_amdgcn_wmma_f32_16x16x4_f32(
            /*neg_a=*/false, a, /*neg_b=*/false, bm,
            /*c_mod=*/(short)0, acc, /*reuse_a=*/false, /*reuse_b=*/false);
    }

    // y = gamma*(w*sim + b - mu)/sqrt(var+eps) + beta, then ReLU
    const float inv   = bn_g[0] * rsqrtf(bn_v[0] + 1e-5f);
    const float scale = inv * w_sim[0];
    const float shift = inv * (b_sim[0] - bn_m[0]) + bn_b[0];

    float* o = out + (size_t)b * NSPIX * NSPIX;
#pragma unroll
    for (int v = 0; v < 8; ++v) {
        const int row = tm * 16 + v + 8 * half;
        const int col = tn * 16 + l;
        const float y = acc[v] * scale + shift;
        o[row * NSPIX + col] = fmaxf(y, 0.0f);
    }
}

// ---------------------------------------------------------------------------
extern "C" void kernel_launch(void* const* d_in, const int* in_sizes, int n_in,
                              void* d_out, int out_size, void* d_ws, size_t ws_size,
                              hipStream_t stream) {
    const float* x      = (const float*)d_in[0];
    const int*   labels = (const int*)d_in[1];
    // d_in[2] = num_spixels (scalar, == NSPIX)
    const float* W_pix  = (const float*)d_in[3];
    const float* b_pix  = (const float*)d_in[4];
    const float* W_red  = (const float*)d_in[5];
    const float* b_red  = (const float*)d_in[6];
    const float* w_sim  = (const float*)d_in[7];
    const float* b_sim  = (const float*)d_in[8];
    const float* bn_g   = (const float*)d_in[9];
    const float* bn_b   = (const float*)d_in[10];
    const float* bn_m   = (const float*)d_in[11];
    const float* bn_v   = (const float*)d_in[12];

    const int B  = 8;
    const int HW = 512 * 512;

    float* mom = (float*)d_ws;                 // B*NSPIX*10 floats
    float* rn  = mom + (size_t)B * NSPIX * 10; // B*NSPIX*FEAT floats

    const int nmom = B * NSPIX * 10;
    zero_kernel<<<(nmom + 255) / 256, 256, 0, stream>>>(mom, nmom);

    moments_kernel<<<dim3(32, B), 256, 0, stream>>>(x, labels, mom, HW);

    segment_stats_kernel<<<B * NSPIX, 64, 0, stream>>>(mom, W_pix, b_pix,
                                                       W_red, b_red, rn);

    const int nwaves = B * 16 * 16;            // one wave per 16x16 tile
    sim_wmma_kernel<<<(nwaves * 32) / 256, 256, 0, stream>>>(
        rn, w_sim, b_sim, bn_g, bn_b, bn_m, bn_v, (float*)d_out);
}